// DirectVoxGO_979252544014
// MI455X (gfx1250) — compile-verified
//
#include <hip/hip_runtime.h>
#include <hip/hip_bf16.h>

// ---------------------------------------------------------------------------
// DirectVoxGO ray marcher for gfx1250 (MI455X).
// One wave32 = one ray; lane = sample index within a 32-sample chunk.
// Gather-bound kernel: grids (65 MB) are L2-resident (192 MB L2), so we
// optimize for gather count (paired b64 z-loads) and latency hiding
// (32768 independent waves, wave-level product scan for compositing).
// ---------------------------------------------------------------------------

#define GRID_N   160
#define GRID_N3  (160 * 160 * 160)
#define NEAR_T   0.2f
#define FAR_T    3.0f
#define STEP_D   0.00625f
#define INV_STEP 160.0f                      // 1 / 0.00625
#define ACT_SHIFT -13.8155095579637739f      // log(1/(1-1e-6) - 1)
#define RAYS_PER_BLOCK 8                     // 256 threads = 8 wave32 rays

#if defined(__HIP_DEVICE_COMPILE__)
typedef __attribute__((address_space(1))) int g_int;   // global
typedef __attribute__((address_space(3))) int l_int;   // LDS
#endif

// Unaligned-safe 8-byte pair load (z0, z0+1 are contiguous): compiler emits
// global_load_b64 when unaligned access is available, else two b32 — both OK.
__device__ __forceinline__ float2 ld2(const float* __restrict__ p) {
  float2 r;
  __builtin_memcpy(&r, p, sizeof(float2));
  return r;
}

// Trilinear interpolation of one channel given 4 (x,y)-row base offsets.
// izb = min(iz0, G-2); zhi means iz0 == G-1, where fz is provably 0, so the
// pair load at izb=(G-2) still yields the correct z0 value in .y.
__device__ __forceinline__ float tri(const float* __restrict__ g,
                                     int r00, int r01, int r10, int r11,
                                     bool zhi, float fx, float fy, float fz) {
  float2 a = ld2(g + r00);
  float2 b = ld2(g + r01);
  float2 c = ld2(g + r10);
  float2 d = ld2(g + r11);
  float a0 = zhi ? a.y : a.x;
  float b0 = zhi ? b.y : b.x;
  float c0 = zhi ? c.y : c.x;
  float d0 = zhi ? d.y : d.x;
  float v00 = fmaf(fz, a.y - a0, a0);
  float v01 = fmaf(fz, b.y - b0, b0);
  float v10 = fmaf(fz, c.y - c0, c0);
  float v11 = fmaf(fz, d.y - d0, d0);
  float v0 = fmaf(fy, v01 - v00, v00);
  float v1 = fmaf(fy, v11 - v10, v10);
  return fmaf(fx, v1 - v0, v0);
}

__global__ void __launch_bounds__(256)
dvgo_render(const float* __restrict__ rays_o,
            const float* __restrict__ rays_d,
            const float* __restrict__ dens,
            const float* __restrict__ k0,
            float* __restrict__ out,
            int nrays) {
  __shared__ float s_ray[RAYS_PER_BLOCK * 8];   // 6 floats/ray, padded to 8

  const int tid  = threadIdx.x;
  const int lane = tid & 31;
  const int wv   = tid >> 5;
  const int ray0 = blockIdx.x * RAYS_PER_BLOCK;

  // ---- Stage per-block ray data into LDS via the gfx1250 async path ----
  if (tid < RAYS_PER_BLOCK * 6) {
    const int r = tid / 6, c = tid % 6;
    const int rr = ray0 + r;
    const float* src = (rr < nrays)
        ? ((c < 3) ? (rays_o + rr * 3 + c) : (rays_d + rr * 3 + (c - 3)))
        : rays_o;
    float* dst = &s_ray[r * 8 + c];
#if defined(__HIP_DEVICE_COMPILE__) && defined(__gfx1250__) && \
    __has_builtin(__builtin_amdgcn_global_load_async_to_lds_b32)
    __builtin_amdgcn_global_load_async_to_lds_b32(
        (g_int*)(void*)const_cast<float*>(src),
        (l_int*)(void*)dst, 0, 0);
#else
    *dst = *src;
#endif
  }
#if defined(__HIP_DEVICE_COMPILE__) && defined(__gfx1250__) && \
    __has_builtin(__builtin_amdgcn_global_load_async_to_lds_b32)
#  if __has_builtin(__builtin_amdgcn_s_wait_asynccnt)
  __builtin_amdgcn_s_wait_asynccnt(0);
#  else
  asm volatile("s_wait_asynccnt 0" ::: "memory");
#  endif
#endif
  __syncthreads();

  const int ray = ray0 + wv;
  if (ray >= nrays) return;

  const float ox = s_ray[wv * 8 + 0], oy = s_ray[wv * 8 + 1], oz = s_ray[wv * 8 + 2];
  const float dx = s_ray[wv * 8 + 3], dy = s_ray[wv * 8 + 4], dz = s_ray[wv * 8 + 5];

  // ---- t range (faithful to reference: divide by max(|d|,1e-6)) ----
  const float avx = fmaxf(fabsf(dx), 1e-6f);
  const float avy = fmaxf(fabsf(dy), 1e-6f);
  const float avz = fmaxf(fabsf(dz), 1e-6f);
  const float t1x = (-1.f - ox) / avx, t2x = (1.f - ox) / avx;
  const float t1y = (-1.f - oy) / avy, t2y = (1.f - oy) / avy;
  const float t1z = (-1.f - oz) / avz, t2z = (1.f - oz) / avz;
  float tmn = fmaxf(fmaxf(fminf(t1x, t2x), fminf(t1y, t2y)), fminf(t1z, t2z));
  float tmx = fminf(fminf(fmaxf(t1x, t2x), fmaxf(t1y, t2y)), fmaxf(t1z, t2z));
  tmn = fminf(fmaxf(tmn, NEAR_T), FAR_T);
  tmx = fminf(fmaxf(tmx, NEAR_T), FAR_T);

  const float nrm = sqrtf(fmaf(dx, dx, fmaf(dy, dy, dz * dz)));
  const int n_steps = (int)fmaxf(ceilf((tmx - tmn) * nrm * INV_STEP), 1.0f);

  const float sx = fmaf(dx, tmn, ox);
  const float sy = fmaf(dy, tmn, oy);
  const float sz = fmaf(dz, tmn, oz);
  const float rn = 1.0f / nrm;
  const float ddx = dx * rn, ddy = dy * rn, ddz = dz * rn;

  const float* __restrict__ k0r = k0;
  const float* __restrict__ k0g = k0 + GRID_N3;
  const float* __restrict__ k0b = k0 + 2 * GRID_N3;

  float Tc = 1.f, accr = 0.f, accg = 0.f, accb = 0.f;

  for (int sbase = 0; sbase < n_steps; sbase += 32) {
    const int   s    = sbase + lane;
    const float dist = (float)s * STEP_D;
    const float px = fmaf(ddx, dist, sx);
    const float py = fmaf(ddy, dist, sy);
    const float pz = fmaf(ddz, dist, sz);

    const bool inb = (s < n_steps) &
        (px >= -1.f) & (px <= 1.f) & (py >= -1.f) & (py <= 1.f) &
        (pz >= -1.f) & (pz <= 1.f);

    float alpha = 0.f, cr = 0.f, cg = 0.f, cb = 0.f;
    if (inb) {
      const float hs = 0.5f * (float)(GRID_N - 1);   // (size-1)/(max-min)
      float ux = fminf(fmaxf((px + 1.f) * hs, 0.f), (float)(GRID_N - 1));
      float uy = fminf(fmaxf((py + 1.f) * hs, 0.f), (float)(GRID_N - 1));
      float uz = fminf(fmaxf((pz + 1.f) * hs, 0.f), (float)(GRID_N - 1));
      const float flx = floorf(ux), fly = floorf(uy), flz = floorf(uz);
      const float fx_ = ux - flx, fy_ = uy - fly, fz_ = uz - flz;
      const int ix0 = (int)flx, iy0 = (int)fly, iz0 = (int)flz;
      const int ix1 = min(ix0 + 1, GRID_N - 1);
      const int iy1 = min(iy0 + 1, GRID_N - 1);
      const int izb = min(iz0, GRID_N - 2);
      const bool zhi = (iz0 == GRID_N - 1);          // implies fz_ == 0
      const int r00 = (ix0 * GRID_N + iy0) * GRID_N + izb;
      const int r01 = (ix0 * GRID_N + iy1) * GRID_N + izb;
      const int r10 = (ix1 * GRID_N + iy0) * GRID_N + izb;
      const int r11 = (ix1 * GRID_N + iy1) * GRID_N + izb;

      const float dv = tri(dens, r00, r01, r10, r11, zhi, fx_, fy_, fz_);
      const float vr = tri(k0r,  r00, r01, r10, r11, zhi, fx_, fy_, fz_);
      const float vg = tri(k0g,  r00, r01, r10, r11, zhi, fx_, fy_, fz_);
      const float vb = tri(k0b,  r00, r01, r10, r11, zhi, fx_, fy_, fz_);

      // alpha = 1 - (1 + exp(d + shift))^(-0.5)  -> v_rsq_f32
      alpha = 1.f - __builtin_amdgcn_rsqf(1.f + __expf(dv + ACT_SHIFT));
      // sigmoid via v_exp + v_rcp
      cr = __builtin_amdgcn_rcpf(1.f + __expf(-vr));
      cg = __builtin_amdgcn_rcpf(1.f + __expf(-vg));
      cb = __builtin_amdgcn_rcpf(1.f + __expf(-vb));
    }

    // ---- wave-wide inclusive product scan of (1 - alpha) (wave32) ----
    float p = 1.f - alpha;
#pragma unroll
    for (int off = 1; off < 32; off <<= 1) {
      float v = __shfl_up(p, off, 32);
      if (lane >= off) p *= v;
    }
    float excl = __shfl_up(p, 1, 32);       // exclusive scan
    if (lane == 0) excl = 1.f;

    const float w = alpha * Tc * excl;      // weight = alpha * T_exclusive
    accr = fmaf(w, cr, accr);
    accg = fmaf(w, cg, accg);
    accb = fmaf(w, cb, accb);

    Tc *= __shfl(p, 31, 32);                // carry full-chunk transmittance
    if (Tc < 1e-7f) break;                  // uniform: remaining weight ~ 0
  }

  // ---- reduce per-lane partial colors across the wave ----
#pragma unroll
  for (int off = 16; off > 0; off >>= 1) {
    accr += __shfl_xor(accr, off, 32);
    accg += __shfl_xor(accg, off, 32);
    accb += __shfl_xor(accb, off, 32);
  }
  if (lane == 0) {
    out[ray * 3 + 0] = accr + Tc;           // white background: + alphainv_last
    out[ray * 3 + 1] = accg + Tc;
    out[ray * 3 + 2] = accb + Tc;
  }
}

extern "C" void kernel_launch(void* const* d_in, const int* in_sizes, int n_in,
                              void* d_out, int out_size, void* d_ws, size_t ws_size,
                              hipStream_t stream) {
  const float* rays_o = (const float*)d_in[0];
  const float* rays_d = (const float*)d_in[1];
  // d_in[2] = xyz_min (= -1), d_in[3] = xyz_max (= +1): baked into constants.
  const float* dens = (const float*)d_in[4];   // [1,160,160,160]
  const float* k0   = (const float*)d_in[5];   // [3,160,160,160]
  float* out = (float*)d_out;

  const int nrays  = in_sizes[0] / 3;
  const int blocks = (nrays + RAYS_PER_BLOCK - 1) / RAYS_PER_BLOCK;
  dvgo_render<<<dim3(blocks), dim3(256), 0, stream>>>(rays_o, rays_d, dens, k0,
                                                      out, nrays);
}